// EPG_X_61529701483107
// MI455X (gfx1250) — compile-verified
//
#include <hip/hip_runtime.h>
#include <hip/hip_bf16.h>
#include <math.h>

// ---------------------------------------------------------------------------
// EPG-X TSE (MT) simulation for MI455X / gfx1250.
//   im=32768 voxels, np2=33 pulses -> kmax=35, K=36 blocks, N=144 state rows.
//   State kept LDS-resident; 4x4 RF transform done with V_WMMA_F32_16X16X4_F32
//   (A = RF matrix in rows 0..3, B = 16 state blocks as columns).
// ---------------------------------------------------------------------------

#define NV        32                 // voxels per workgroup
#define KBLK      36                 // phase-order blocks per voxel (kmax+1)
#define NROW      144                // 4*KBLK state rows per voxel
#define NBLK      (NV * KBLK)        // 1152 flat blocks per workgroup
#define NTIL      (NBLK / 16)        // 72 WMMA tiles per workgroup
#define NSTEP     32                 // scan steps (np2-1)
#define NTHREADS  256
#define NWAVES    (NTHREADS / 32)
#define KMAXB     35                 // last block index

// LDS layout (floats)
#define OFF_FF    0                        // FF state  [NV*NROW]
#define OFF_FC    (NV * NROW)              // Fcol      [NV*NROW]
#define OFF_CST   (2 * NV * NROW)          // per-voxel consts [NV*8]
#define OFF_AREF  (OFF_CST + NV * 8)       // 4x4 refoc RF matrix [16]
#define LDS_FLOATS (OFF_AREF + 16)

typedef float v2f __attribute__((ext_vector_type(2)));
typedef float v8f __attribute__((ext_vector_type(8)));

#if defined(__HIP_DEVICE_COMPILE__) && __has_builtin(__builtin_amdgcn_wmma_f32_16x16x4_f32)
#define HAVE_WMMA_F32_16X16X4 1
#else
#define HAVE_WMMA_F32_16X16X4 0
#endif

// relax + shift + bvec for all blocks; optionally stream step outputs.
//   q = active block count (m = 4q rows).  zeroInactive: write zeros for b>=q.
__device__ __forceinline__ void relax_shift_phase(
    float* S, int srcOff, int dstOff, int q, bool zeroInactive,
    bool doOut, int s, int vg0,
    float* __restrict__ outF0, float* __restrict__ outFn,
    float* __restrict__ outZn)
{
    for (int g = threadIdx.x; g < NBLK; g += NTHREADS) {
        const int v = g / KBLK;
        const int b = g - v * KBLK;
        const float* src = &S[srcOff + v * NROW];
        float*       dst = &S[dstOff + v * NROW + b * 4];
        const float* cst = &S[OFF_CST + v * 8];
        float y0 = 0.f, y1 = 0.f, y2 = 0.f, y3 = 0.f;
        if (b < q) {
            const float E2 = cst[0];
            // F+ : shift down one order (block 0 takes conj path from row 5)
            y0 = E2 * ((b == 0) ? src[5] : src[(b - 1) * 4]);
            // F- : shift up one order; zero past kmax or past row mask
            if (b < KMAXB && (b + 1) < q) y1 = E2 * src[(b + 1) * 4 + 1];
            // Z  : 2x2 exchange-relaxation (XiL), + recovery offset at b==0
            const float za = src[b * 4 + 2], zb = src[b * 4 + 3];
            y2 = cst[1] * za + cst[2] * zb;
            y3 = cst[3] * za + cst[4] * zb;
            if (b == 0) { y2 += cst[5]; y3 += cst[6]; }
            dst[0] = y0; dst[1] = y1; dst[2] = y2; dst[3] = y3;
        } else if (zeroInactive) {
            dst[0] = 0.f; dst[1] = 0.f; dst[2] = 0.f; dst[3] = 0.f;
        }
        if (doOut) {   // Fseq -> (F0, Fn, Zn); zeros for inactive blocks too
            const int vg = vg0 + v;
            if (b == 0) {
                outF0[(size_t)vg * NSTEP + s] = y0;
                // row 0 also appears inside Fn at j == 33 (idx list dup)
                outFn[((size_t)vg * 69 + 33) * NSTEP + s] = y0;
            } else {
                outFn[((size_t)vg * 69 + 33 + b) * NSTEP + s] = y0;
            }
            if (b >= 1 && b <= 33)
                outFn[((size_t)vg * 69 + 33 - b) * NSTEP + s] = y1;
            float* zp = &outZn[(((size_t)vg * KBLK + b) * NSTEP + s) * 2];
            zp[0] = y2; zp[1] = y3;
        }
    }
}

// Apply 4x4 refocusing RF matrix to blocks b<q of FF via WMMA 16x16x4 f32.
//   A (16x4): rows 0..3 = Aref, rows 4..15 zero. Lane=M, VGPR pair = K.
//   B (4x16): column n = 4-vector of block (tile*16+n). Lane=N, VGPR pair = K.
//   D (16x16): rows 0..3 = transformed blocks -> lanes 0..15 write d[0..3].
__device__ __forceinline__ void rf_wmma_phase(float* S, int q, v2f aFrag,
                                              int lane, int kbase)
{
    const int wave = threadIdx.x >> 5;
    const int col  = lane & 15;
    for (int t = wave; t < NTIL; t += NWAVES) {
        const int blk = t * 16 + col;
#if HAVE_WMMA_F32_16X16X4
        const float* p = &S[OFF_FF + blk * 4 + kbase];
        v2f bFrag;
        bFrag.x = p[0];
        bFrag.y = p[1];
        v8f d = {0.f, 0.f, 0.f, 0.f, 0.f, 0.f, 0.f, 0.f};
        // D = A x B + 0   (8-arg f32 WMMA form)
        d = __builtin_amdgcn_wmma_f32_16x16x4_f32(
                /*neg_a=*/false, aFrag, /*neg_b=*/false, bFrag,
                /*c_mod=*/(short)0, d, /*reuse_a=*/false, /*reuse_b=*/false);
        if (lane < 16) {
            const int b = blk % KBLK;
            if (b < q) {
                float* o = &S[OFF_FF + blk * 4];
                o[0] = d[0]; o[1] = d[1]; o[2] = d[2]; o[3] = d[3];
            }
        }
#else
        (void)aFrag; (void)kbase;
        if (lane < 16) {
            const int b = blk % KBLK;
            if (b < q) {
                float* x = &S[OFF_FF + blk * 4];
                const float* A = &S[OFF_AREF];
                const float x0 = x[0], x1 = x[1], x2 = x[2], x3 = x[3];
                const float r0 = A[0]*x0  + A[1]*x1  + A[2]*x2  + A[3]*x3;
                const float r1 = A[4]*x0  + A[5]*x1  + A[6]*x2  + A[7]*x3;
                const float r2 = A[8]*x0  + A[9]*x1  + A[10]*x2 + A[11]*x3;
                const float r3 = A[12]*x0 + A[13]*x1 + A[14]*x2 + A[15]*x3;
                x[0] = r0; x[1] = r1; x[2] = r2; x[3] = r3;
            }
        }
#endif
    }
}

__global__ __launch_bounds__(NTHREADS)
void epgx_tse_mt_kernel(
    const float* __restrict__ theta, const float* __restrict__ b1t,
    const float* __restrict__ R1a,   const float* __restrict__ R2a,
    const float* __restrict__ f,     const float* __restrict__ ka,
    const float* __restrict__ kb,    const float* __restrict__ z0,
    const int* __restrict__ espP,    const int* __restrict__ gP,
    float* __restrict__ outF0, float* __restrict__ outFn,
    float* __restrict__ outZn)
{
    __shared__ float S[LDS_FLOATS];
    const int tid = threadIdx.x;
    const int vg0 = blockIdx.x * NV;

    const float ESP = (float)espP[0];
    const float h   = 0.5f * ESP;
    const float Gms = (float)gP[0] * 0.001f;
    const float wtScale =
        3.14159265358979323846f * (0.2675221f * 0.2675221f) * Gms;

    // ---- zero state buffers; thread 0 builds the refocusing RF matrix
    for (int i = tid; i < 2 * NV * NROW; i += NTHREADS) S[i] = 0.f;
    if (tid == 0) {
        const float al  = fabsf(theta[1]);
        const float wt1 = wtScale * b1t[1];
        const float chh = cosf(al * 0.5f), shh = sinf(al * 0.5f);
        const float c2 = chh * chh, s2 = shh * shh;
        const float c = cosf(al), sn = sinf(al);
        const float e = expf(-wt1);
        float* A = &S[OFF_AREF];
        A[0]  = c2;        A[1]  = s2;       A[2]  = sn;  A[3]  = 0.f;
        A[4]  = s2;        A[5]  = c2;       A[6]  = -sn; A[7]  = 0.f;
        A[8]  = -0.5f*sn;  A[9]  = 0.5f*sn;  A[10] = c;   A[11] = 0.f;
        A[12] = 0.f;       A[13] = 0.f;      A[14] = 0.f; A[15] = e;
    }
    __syncthreads();

    // ---- per-voxel constants (E2, XiL, Zoff) + excitation of block 0
    if (tid < NV) {
        const int v  = tid;
        const int vg = vg0 + v;
        const float r1  = R1a[vg], r2v = R2a[vg], fv = f[vg];
        const float kav = ka[vg],  kbv = kb[vg];
        float* cst = &S[OFF_CST + v * 8];

        const float E2    = expf(-h * r2v);
        const float ksum  = kav + kbv;
        const float delta = h * 0.5f * ksum;            // > 0
        const float et    = expf(-h * (r1 + 0.5f * ksum));
        const float chd   = coshf(delta);
        const float sd    = sinhf(delta) / delta;
        const float X00 = et * (chd + sd * h * 0.5f * (kbv - kav));
        const float X01 = et * (sd * h * kbv);
        const float X10 = et * (sd * h * kav);
        const float X11 = et * (chd + sd * h * 0.5f * (kav - kbv));
        // Zoff = (XiL - I) * Lam^-1 * CL
        const float det = r1 * r1 + r1 * ksum;
        const float cl0 = (1.f - fv) * r1;
        const float cl1 = fv * r1;
        const float w0  = ((-r1 - kbv) * cl0 - kbv * cl1) / det;
        const float w1  = (-kav * cl0 + (-r1 - kav) * cl1) / det;
        cst[0] = E2;
        cst[1] = X00; cst[2] = X01; cst[3] = X10; cst[4] = X11;
        cst[5] = (X00 - 1.f) * w0 + X01 * w1;
        cst[6] = X10 * w0 + (X11 - 1.f) * w1;
        cst[7] = 0.f;
        // FF[:, :4] = rf_ex(|theta0|, WT0) @ [0,0,za,zb]
        const float za = z0[vg * 2 + 0], zb = z0[vg * 2 + 1];
        const float al0 = fabsf(theta[0]);
        const float s0 = sinf(al0), c0 = cosf(al0);
        const float e0 = expf(-wtScale * b1t[0]);
        float* ff0 = &S[OFF_FF + v * NROW];
        ff0[0] = s0 * za;
        ff0[1] = s0 * za;
        ff0[2] = c0 * za;
        ff0[3] = e0 * zb;
    }
    __syncthreads();

    // ---- per-lane constant A fragment for WMMA (Aref in rows 0..3)
    const int lane  = tid & 31;
    const int kbase = (lane < 16) ? 0 : 2;
    v2f aFrag;
    {
        const int row = lane & 15;
        if (row < 4) {
            aFrag.x = S[OFF_AREF + row * 4 + kbase];
            aFrag.y = S[OFF_AREF + row * 4 + kbase + 1];
        } else {
            aFrag.x = 0.f;
            aFrag.y = 0.f;
        }
    }

    // ---- FF = shift(relax(FF), N) + bvec     (full mask, m = N)
    relax_shift_phase(S, OFF_FF, OFF_FC, KBLK, true, false, 0, vg0,
                      nullptr, nullptr, nullptr);
    __syncthreads();
    for (int i = tid; i < NV * NROW; i += NTHREADS) S[OFF_FF + i] = S[OFF_FC + i];
    __syncthreads();

    // ---- echo-train scan: m_arr = 4*kpp[s+1]
    for (int s = 0; s < NSTEP; ++s) {
        const int q = (s < 16) ? (2 * s + 5) : (65 - 2 * s);  // kpp[s+1]
        // FF = where(rmask, Aref @ FF, FF)
        rf_wmma_phase(S, q, aFrag, lane, kbase);
        __syncthreads();
        // Fcol = where(rmask, shift(relax(FF), m) + bvec, 0) ; stream outputs
        relax_shift_phase(S, OFF_FF, OFF_FC, q, true, true, s, vg0,
                          outF0, outFn, outZn);
        __syncthreads();
        // FF = where(rmask, shift(relax(Fcol), m) + bvec, FF)
        relax_shift_phase(S, OFF_FC, OFF_FF, q, false, false, 0, vg0,
                          nullptr, nullptr, nullptr);
        __syncthreads();
    }
}

extern "C" void kernel_launch(void* const* d_in, const int* in_sizes, int n_in,
                              void* d_out, int out_size, void* d_ws,
                              size_t ws_size, hipStream_t stream)
{
    (void)n_in; (void)out_size; (void)d_ws; (void)ws_size;
    const float* theta = (const float*)d_in[0];
    const float* b1t   = (const float*)d_in[1];
    const float* R1a   = (const float*)d_in[2];
    const float* R2a   = (const float*)d_in[3];
    const float* f     = (const float*)d_in[4];
    const float* ka    = (const float*)d_in[5];
    const float* kb    = (const float*)d_in[6];
    const float* z0    = (const float*)d_in[7];
    const int*   esp   = (const int*)d_in[8];
    const int*   g     = (const int*)d_in[9];

    const int im = in_sizes[2];               // R1a flat size == #voxels
    float* out   = (float*)d_out;
    float* outF0 = out;                                       // (im, 32)
    float* outFn = out + (size_t)im * NSTEP;                  // (im, 69, 32)
    float* outZn = outFn + (size_t)im * 69 * NSTEP;           // (im, 36, 32, 2)

    dim3 grid(im / NV), block(NTHREADS);
    hipLaunchKernelGGL(epgx_tse_mt_kernel, grid, block, 0, stream,
                       theta, b1t, R1a, R2a, f, ka, kb, z0, esp, g,
                       outF0, outFn, outZn);
}